// HeatmapGenerator_7146825580723
// MI455X (gfx1250) — compile-verified
//
#include <hip/hip_runtime.h>

// Problem shape (fixed by the harness / reference setup_inputs):
//   heatmap [B=64, K=17, H=256, W=256] f32, window [11,11] f32,
//   keypoints [B=64, A=8, K=17, 3] i32 (x, y, vis)
#define B_DIM 64
#define A_DIM 8
#define K_DIM 17
#define H_DIM 256
#define W_DIM 256
#define WS_DIM 11
#define HW_DIM 5

// ---------------------------------------------------------------------------
// Kernel 1: base-heatmap propagation, out = in. Pure streaming copy.
// Each thread copies 64 contiguous bytes (4 consecutive float4s) so LLVM
// clusters 4 global_load_b128 back-to-back (16 live VGPRs) before the first
// s_wait_loadcnt, then issues 4 global_store_b128. Grid is sized so the main
// loop runs exactly once per thread. ~570 MB HBM traffic ~= 24 us @ 23.3 TB/s.
// ---------------------------------------------------------------------------
__global__ __launch_bounds__(256) void hm_copy_b128x4(const float4* __restrict__ in,
                                                      float4* __restrict__ out,
                                                      int n4) {
  const int step = gridDim.x * blockDim.x * 4;
  int i = (blockIdx.x * blockDim.x + threadIdx.x) * 4;

  // Main loop: 4 consecutive b128 loads in flight, then 4 b128 stores.
  for (; i + 3 < n4; i += step) {
    float4 a = in[i + 0];
    float4 b = in[i + 1];
    float4 c = in[i + 2];
    float4 d = in[i + 3];
    out[i + 0] = a;
    out[i + 1] = b;
    out[i + 2] = c;
    out[i + 3] = d;
  }
  // Tail (not taken for the harness shape, kept for robustness).
  for (; i < n4; ++i) {
    out[i] = in[i];
  }
}

// ---------------------------------------------------------------------------
// Kernel 2: scatter-max of the 11x11 Gaussian window for each keypoint.
// One workgroup per keypoint (8704 total); threads 0..120 cover the window.
// Overlapping windows are resolved with the CDNA5 native no-return float-max
// atomic GLOBAL_ATOMIC_MAX_NUM_F32 (IEEE maximumNumber, matches jnp max;
// tracked via STOREcnt, implicitly drained at s_endpgm).
// ---------------------------------------------------------------------------
__device__ __forceinline__ void atomic_max_num_f32(float* addr, float v) {
  asm volatile("global_atomic_max_num_f32 %0, %1, off"
               :
               : "v"(addr), "v"(v)
               : "memory");
}

__global__ __launch_bounds__(128) void hm_scatter(const int* __restrict__ kp,
                                                  const float* __restrict__ win,
                                                  float* __restrict__ hm) {
  const int bk = blockIdx.x;              // flat (b, a, k) in [0, B*A*K)
  // Wave-uniform keypoint fields -> scalar loads (KMcnt path).
  const int x   = kp[bk * 3 + 0];
  const int y   = kp[bk * 3 + 1];
  const int vis = kp[bk * 3 + 2];
  if (vis < 1) return;                    // keypoint not labeled: no-op

  const int t = threadIdx.x;
  if (t >= WS_DIM * WS_DIM) return;       // 121 active lanes of 128

  const int wy  = t / WS_DIM;
  const int wx  = t - wy * WS_DIM;
  const int row = y + wy - HW_DIM;
  const int col = x + wx - HW_DIM;
  if ((unsigned)row >= (unsigned)H_DIM) return;   // clip to image
  if ((unsigned)col >= (unsigned)W_DIM) return;

  const int b = bk / (A_DIM * K_DIM);
  const int k = bk % K_DIM;

  const float v = win[t];                 // window[wy][wx], cache-resident
  const long idx = ((long)((b * K_DIM + k) * H_DIM + row)) * W_DIM + col;
  atomic_max_num_f32(hm + idx, v);
}

// ---------------------------------------------------------------------------
// Launch
// ---------------------------------------------------------------------------
extern "C" void kernel_launch(void* const* d_in, const int* in_sizes, int n_in,
                              void* d_out, int out_size, void* d_ws, size_t ws_size,
                              hipStream_t stream) {
  const float* hm_in = (const float*)d_in[0];   // [B,K,H,W] f32 (base heatmap)
  const float* win   = (const float*)d_in[1];   // [11,11]   f32
  const int*   kp    = (const int*)d_in[2];     // [B,A,K,3] i32
  float*       out   = (float*)d_out;           // [B,K,H,W] f32

  // 1) out = base heatmap. 64 B per thread, one main iteration per thread.
  const int n4 = out_size / 4;                  // 17,825,792 float4's
  const int threads = 256;
  const int blocks = (n4 / 4 + threads - 1) / threads;  // 17,408
  hm_copy_b128x4<<<blocks, threads, 0, stream>>>((const float4*)hm_in,
                                                 (float4*)out, n4);

  // 2) scatter-max the Gaussian windows (8704 keypoints, 121 px each).
  const int nkp = in_sizes[2] / 3;              // B*A*K = 8704
  hm_scatter<<<nkp, 128, 0, stream>>>(kp, win, out);
}